// myNN_7937099563073
// MI455X (gfx1250) — compile-verified
//
#include <hip/hip_runtime.h>
#include <stdint.h>

// out[r,:] = Wt[x1[r],:] + Wt[x2[r]+256,:] + b      (B=524288 rows, 256 cols, f32)
// Memory-bound: 512MB NT stores dominate (~22us floor @ 23.3 TB/s).
// CDNA5 paths used: TDM tensor_load_to_lds (stage 256KB half-table to LDS),
// async global->LDS (bias), s_wait_tensorcnt / s_wait_asynccnt, wave32 b128 streaming.

#define VOCABSZ 256
#define ROWF    256   // f32 per output row

typedef float    vf4 __attribute__((ext_vector_type(4)));
typedef uint32_t v4u __attribute__((ext_vector_type(4)));
typedef uint32_t v8u __attribute__((ext_vector_type(8)));

__global__ __launch_bounds__(1024) void emb2_fused_kernel(
    const int* __restrict__ x,        // (B,2) int32
    const float* __restrict__ Wt,     // (512,256) f32
    const float* __restrict__ bias,   // (256,)   f32
    float* __restrict__ out,          // (B,256)  f32
    int B, int rowsPerBlock)
{
    __shared__ __align__(16) float ltab[VOCABSZ * ROWF]; // 256 KB: Wt rows [0,256)
    __shared__ __align__(16) float lbias[ROWF];          // 1 KB

    const int tid  = threadIdx.x;
    const int wave = tid >> 5;
    const int lane = tid & 31;

    // ---- Stage Wt[0:256,:] into LDS with a single TDM op (wave 0 only) ----
    if (wave == 0) {
        const uint64_t ga     = (uint64_t)(uintptr_t)Wt;
        const uint32_t ldsoff = (uint32_t)(uintptr_t)(&ltab[0]); // LDS byte offset
        v4u g0;
        g0[0] = 1u;                                   // count=1, user descriptor
        g0[1] = ldsoff;                               // lds_addr (bytes)
        g0[2] = (uint32_t)(ga & 0xffffffffu);         // global_addr[31:0]
        g0[3] = (uint32_t)((ga >> 32) & 0x01ffffffu)  // global_addr[56:32]
              | (2u << 30);                           // type=2 ("image")
        v8u g1;
        g1[0] = (2u << 16);           // data_size=2 -> 4-byte elements; no multicast
        g1[1] = ((uint32_t)ROWF    << 16);  // tensor_dim0[15:0] = 256   @ bits 63:48
        g1[2] = ((uint32_t)VOCABSZ << 16);  // tensor_dim1[15:0] = 256   @ bits 95:80
        g1[3] = ((uint32_t)ROWF    << 16);  // tile_dim0 = 256           @ bits 127:112
        g1[4] = (uint32_t)VOCABSZ;          // tile_dim1 = 256, tile_dim2 = 0
        g1[5] = (uint32_t)ROWF;             // tensor_dim0_stride = 256 elements
        g1[6] = 0u;                         // stride hi / dim1_stride = 0 (unused, 2D)
        g1[7] = 0u;
        asm volatile("tensor_load_to_lds %0, %1" :: "s"(g0), "s"(g1) : "memory");
        __builtin_amdgcn_s_wait_tensorcnt(0);
    }

    // ---- Stage bias (1KB) into LDS via async global->LDS b128 (waves 1..2) ----
    if (tid >= 32 && tid < 96) {
        const uint32_t t      = (uint32_t)(tid - 32);          // 0..63, 16B each
        const uint32_t ldsoff = (uint32_t)(uintptr_t)(&lbias[0]) + t * 16u;
        const uint32_t goff   = t * 16u;
        const uint64_t gb     = (uint64_t)(uintptr_t)bias;
        asm volatile("global_load_async_to_lds_b128 %0, %1, %2"
                     :: "v"(ldsoff), "v"(goff), "s"(gb) : "memory");
        asm volatile("s_wait_asynccnt 0" ::: "memory");
    }
    __syncthreads();

    // Per-lane fixed columns: lane covers [4*lane,4*lane+4) and +128.
    const int c0 = lane * 4;
    const int c1 = c0 + 128;
    const vf4 bv0 = *(const vf4*)(lbias + c0);   // bias kept in registers
    const vf4 bv1 = *(const vf4*)(lbias + c1);

    const int rbeg = blockIdx.x * rowsPerBlock;
    int rend = rbeg + rowsPerBlock;
    if (rend > B) rend = B;

    // One row per wave per iteration; 32 waves per block.
    for (int r = rbeg + wave; r < rend; r += 32) {
        const int2 xi = ((const int2*)x)[r];                 // wave-uniform address
        const float* t1 = &ltab[(uint32_t)xi.x * ROWF];      // LDS gather (x1)
        const float* t2 = Wt + ((size_t)xi.y + VOCABSZ) * ROWF; // L2 gather (x2+256)
        float* o = out + (size_t)r * ROWF;

        vf4 a0 = *(const vf4*)(t1 + c0);      // ds_load_b128
        vf4 a1 = *(const vf4*)(t1 + c1);
        vf4 g0v = *(const vf4*)(t2 + c0);     // global_load_b128 (L2-resident)
        vf4 g1v = *(const vf4*)(t2 + c1);

        vf4 r0 = a0 + g0v + bv0;
        vf4 r1 = a1 + g1v + bv1;

        __builtin_nontemporal_store(r0, (vf4*)(o + c0));  // streaming NT b128 stores
        __builtin_nontemporal_store(r1, (vf4*)(o + c1));
    }
}

extern "C" void kernel_launch(void* const* d_in, const int* in_sizes, int n_in,
                              void* d_out, int out_size, void* d_ws, size_t ws_size,
                              hipStream_t stream) {
    const int*   x    = (const int*)d_in[0];
    const float* Wt   = (const float*)d_in[1];
    const float* bias = (const float*)d_in[2];
    float*       out  = (float*)d_out;

    const int B = in_sizes[0] / 2;           // (B,2) indices
    const int grid = 512;                    // 257KB LDS -> 1 block/WGP; B/512 = 1024 rows each
    const int rowsPerBlock = (B + grid - 1) / grid;

    emb2_fused_kernel<<<grid, 1024, 0, stream>>>(x, Wt, bias, out, B, rowsPerBlock);
}